// WaveletTransform3D_33698313404648
// MI455X (gfx1250) — compile-verified
//
#include <hip/hip_runtime.h>
#include <cstdint>

// 3D Haar LL band over 2x2x2 blocks:
//   out[b,c,od,oh,ow] = sum(x[b,c,2od:2od+2, 2oh:2oh+2, 2ow:2ow+2]) * 1/(2*sqrt(2))
// Shapes fixed by the reference: x (2,32,64,128,128) f32 -> out (2,32,32,64,64) f32.
// Pure streaming kernel (~302 MB @ 23.3 TB/s ~= 13 us). Data movement uses the
// CDNA5 async global->LDS path (GLOBAL_LOAD_ASYNC_TO_LDS_B128, ASYNCcnt).

#define HAAR_LL_SCALE 0.35355339059327373f

namespace {
constexpr int B  = 2,  C = 32, D = 64, H = 128, W = 128;
constexpr int BC = B * C;              // 64
constexpr int Do = D / 2;              // 32
constexpr int Ho = H / 2;              // 64
constexpr int Wo = W / 2;              // 64
constexpr int HW  = H * W;             // 16384
constexpr int DHW = D * HW;            // 1048576

// Per-workgroup tile: input 2(D) x 8(H) x 128(W) f32 = 8 KB in LDS,
// producing a 1(D) x 4(H) x 64(W) output tile (256 outputs, 1 per thread).
constexpr int TILE_FLOATS = 2 * 8 * W;     // 2048 floats
constexpr int TILE_CHUNKS = TILE_FLOATS/4; // 512 x b128 chunks
constexpr int BLOCKS = BC * Do * (Ho / 4); // 64*32*16 = 32768
} // namespace

__global__ __launch_bounds__(256)
void haar_ll3d_async_kernel(const float* __restrict__ x, float* __restrict__ out) {
    __shared__ float smem[TILE_FLOATS];  // [d(2)][h(8)][w(128)] contiguous

    const int t = threadIdx.x;
    const unsigned blk = blockIdx.x;
    const int ohBlk = blk & 15;          // 16 groups of 4 output rows
    const int od    = (blk >> 4) & 31;   // 32 output depths
    const int bc    = blk >> 9;          // 64 batch*channel slabs

    // Base of this workgroup's 2x8x128 input tile.
    const float* src = x + (size_t)bc * DHW + (size_t)(2 * od) * HW
                         + (size_t)(8 * ohBlk) * W;

    // Low 32 bits of a generic LDS pointer == wave-relative LDS byte offset
    // (shared aperture lives in addr[63:32]).
    const unsigned ldsBase = (unsigned)(uintptr_t)(&smem[0]);

    // Stage-in: 512 16-byte chunks, 2 per thread; 16B-aligned, coalesced
    // (32 lanes * 16 B = 512 B = exactly one W row per wave-instruction).
#pragma unroll
    for (int k = 0; k < 2; ++k) {
        const int c   = t + 256 * k;       // chunk id in [0, 512)
        const int w4  = c & 31;            // float4 column within row
        const int row = c >> 5;            // 0..15: d = row>>3, h = row&7
        const float* g = src + (size_t)(row >> 3) * HW
                             + (size_t)(row & 7) * W + (size_t)(w4 * 4);
        const unsigned ldsOff = ldsBase + (unsigned)c * 16u;
        asm volatile("global_load_async_to_lds_b128 %0, %1, off"
                     :: "v"(ldsOff), "v"(g)
                     : "memory");
    }
    // Wait for this wave's async copies to land in LDS, then make all waves'
    // portions visible to the whole workgroup.
    asm volatile("s_wait_asynccnt 0" ::: "memory");
    __syncthreads();

    // Compute: one output per thread. Conflict-free float2 LDS reads.
    const int oh_l = t >> 6;             // 0..3 local output row
    const int ow   = t & 63;             // 0..63 output column
    const int h0   = 2 * oh_l;

    float s = 0.0f;
#pragma unroll
    for (int d = 0; d < 2; ++d) {
#pragma unroll
        for (int hh = 0; hh < 2; ++hh) {
            const int base = ((d << 3) + h0 + hh) * W + 2 * ow;
            s += smem[base] + smem[base + 1];
        }
    }

    const size_t oidx = (((size_t)bc * Do + od) * Ho + (size_t)(4 * ohBlk + oh_l)) * Wo + ow;
    out[oidx] = s * HAAR_LL_SCALE;
}

extern "C" void kernel_launch(void* const* d_in, const int* in_sizes, int n_in,
                              void* d_out, int out_size, void* d_ws, size_t ws_size,
                              hipStream_t stream) {
    (void)in_sizes; (void)n_in; (void)out_size; (void)d_ws; (void)ws_size;
    const float* x = (const float*)d_in[0];
    float* out = (float*)d_out;
    haar_ll3d_async_kernel<<<dim3(BLOCKS), dim3(256), 0, stream>>>(x, out);
}